// PretrainGIN_75076028334404
// MI455X (gfx1250) — compile-verified
//
#include <hip/hip_runtime.h>

#define D       128
#define TILE_M  64
#define ZS      132   // padded LDS row stride (floats) for activation tiles
#define WS      132   // padded LDS row stride (floats) for weight tile
#define WCS     20    // padded LDS row stride for classifier weight

typedef __attribute__((ext_vector_type(2))) float v2f;
typedef __attribute__((ext_vector_type(8))) float v8f;

__device__ __forceinline__ v8f wmma_f32_4(v2f a, v2f b, v8f c) {
    // D = A(16x4,f32) * B(4x16,f32) + C(16x16,f32)  — full f32 precision matrix op
    return __builtin_amdgcn_wmma_f32_16x16x4_f32(false, a, false, b, (short)0, c,
                                                 false, false);
}

// ============================ CSR construction (edge list is layer-invariant) =
__global__ void zero_i32(int* __restrict__ p, int n) {
    int i = blockIdx.x * blockDim.x + threadIdx.x;
    if (i < n) p[i] = 0;
}

__global__ void count_deg(const int* __restrict__ dst, int* __restrict__ deg, int ne) {
    int e = blockIdx.x * blockDim.x + threadIdx.x;
    if (e < ne) atomicAdd(&deg[dst[e]], 1);
}

// Single-block exclusive scan of deg[0..n) -> rowptr[0..n]; mirrors into cursor.
// NOTE: deg may alias rowptr (in-place), so read deg[i] before overwriting.
__global__ __launch_bounds__(256) void scan_rowptr(const int* __restrict__ deg,
                                                   int* __restrict__ rowptr,
                                                   int* __restrict__ cursor, int n) {
    __shared__ int sums[256];
    const int tid = threadIdx.x;
    const int chunk = (n + 255) / 256;
    const int lo = tid * chunk;
    const int hi = min(lo + chunk, n);
    int s = 0;
    for (int i = lo; i < hi; ++i) s += deg[i];
    sums[tid] = s;
    __syncthreads();
    if (tid == 0) {
        int acc = 0;
        for (int t = 0; t < 256; ++t) { int v = sums[t]; sums[t] = acc; acc += v; }
    }
    __syncthreads();
    int acc = sums[tid];
    for (int i = lo; i < hi; ++i) {
        int dv = deg[i];        // read BEFORE overwrite (deg aliases rowptr)
        rowptr[i] = acc;
        cursor[i] = acc;
        acc += dv;
    }
    if (tid == 255) rowptr[n] = acc;
}

__global__ void fill_csr(const int* __restrict__ src, const int* __restrict__ dst,
                         int* __restrict__ cursor, int* __restrict__ col, int ne) {
    int e = blockIdx.x * blockDim.x + threadIdx.x;
    if (e < ne) {
        int pos = atomicAdd(&cursor[dst[e]], 1);
        col[pos] = src[e];
    }
}

// ---------------- z[i] = h[i] + sum_{j in csr row i} h[col[j]]  (no atomics) --
// One wave32 per node; each lane owns a float4 column slice (32*4 = 128 = D).
__global__ void gin_aggregate(const float* __restrict__ h, const int* __restrict__ rowptr,
                              const int* __restrict__ col, float* __restrict__ z,
                              int n_nodes) {
    int node = (blockIdx.x * blockDim.x + threadIdx.x) >> 5;
    if (node >= n_nodes) return;
    int lane = threadIdx.x & 31;
    const float4* hv = (const float4*)h;
    float4 acc = hv[(size_t)node * 32 + lane];          // self term: (1+eps)*x_i, eps=0
    int beg = rowptr[node];
    int end = rowptr[node + 1];
    for (int j = beg; j < end; ++j) {
        int s = col[j];
        float4 v = hv[(size_t)s * 32 + lane];
        acc.x += v.x; acc.y += v.y; acc.z += v.z; acc.w += v.w;
    }
    ((float4*)z)[(size_t)node * 32 + lane] = acc;
}

// ===================== fallback path when workspace can't hold CSR ============
__global__ void copy_f4(const float4* __restrict__ in, float4* __restrict__ out, int n4) {
    int i = blockIdx.x * blockDim.x + threadIdx.x;
    if (i < n4) out[i] = in[i];
}

__global__ void gin_scatter(const float* __restrict__ h, const int* __restrict__ src,
                            const int* __restrict__ dst, float* __restrict__ z,
                            int num_edges) {
    int e = (blockIdx.x * blockDim.x + threadIdx.x) >> 5;
    if (e >= num_edges) return;
    int lane = threadIdx.x & 31;
    int s = src[e];
    int d = dst[e];
    const float4 v = *reinterpret_cast<const float4*>(h + (size_t)s * D + lane * 4);
    float* zp = z + (size_t)d * D + lane * 4;
    unsafeAtomicAdd(zp + 0, v.x);
    unsafeAtomicAdd(zp + 1, v.y);
    unsafeAtomicAdd(zp + 2, v.z);
    unsafeAtomicAdd(zp + 3, v.w);
}

// ==================== out = relu(z@W1+b1)@W2+b2, fused per 64-row tile ========
__global__ __launch_bounds__(256) void gin_mlp(const float* __restrict__ z,
                                               const float* __restrict__ W1,
                                               const float* __restrict__ b1,
                                               const float* __restrict__ W2,
                                               const float* __restrict__ b2,
                                               float* __restrict__ out, int n_nodes) {
    __shared__ float z_lds[TILE_M * ZS];
    __shared__ float w_lds[D * WS];
    __shared__ float h_lds[TILE_M * ZS];

    const int tid  = threadIdx.x;
    const int wave = tid >> 5;       // 8 waves: each owns a 16-col N strip
    const int lane = tid & 31;
    const int ln15 = lane & 15;
    const int kh   = lane >> 4;      // which K-half of the 16x4 / 4x16 fragment
    const int m_block = blockIdx.x * TILE_M;

    // Stage z tile (64x128) into LDS, zero-padding rows past n_nodes.
    for (int i = tid; i < TILE_M * (D / 4); i += 256) {
        int row = i / (D / 4), col4 = i % (D / 4);
        float4 v = make_float4(0.f, 0.f, 0.f, 0.f);
        int gr = m_block + row;
        if (gr < n_nodes) v = *reinterpret_cast<const float4*>(z + (size_t)gr * D + col4 * 4);
        *reinterpret_cast<float4*>(&z_lds[row * ZS + col4 * 4]) = v;
    }
    // Stage W1 (128x128) into LDS.
    for (int i = tid; i < D * (D / 4); i += 256) {
        int row = i / (D / 4), col4 = i % (D / 4);
        float4 v = *reinterpret_cast<const float4*>(W1 + row * D + col4 * 4);
        *reinterpret_cast<float4*>(&w_lds[row * WS + col4 * 4]) = v;
    }
    __syncthreads();

    // GEMM1: hidden = relu(z @ W1 + b1)
    {
        const int n0 = wave * 16;
        const float bias = b1[n0 + ln15];
        for (int mt = 0; mt < 4; ++mt) {
            v8f acc = {};
            const float* arow = &z_lds[(mt * 16 + ln15) * ZS + 2 * kh];
            const float* brow = &w_lds[(2 * kh) * WS + n0 + ln15];
            for (int k = 0; k < D; k += 4) {
                v2f a, b;
                a.x = arow[k];          a.y = arow[k + 1];
                b.x = brow[k * WS];     b.y = brow[k * WS + WS];
                acc = wmma_f32_4(a, b, acc);
            }
            for (int r = 0; r < 8; ++r) {
                int row = mt * 16 + r + kh * 8;          // C/D layout: M = r + 8*(lane>=16)
                float v = acc[r] + bias;
                h_lds[row * ZS + n0 + ln15] = v > 0.f ? v : 0.f;
            }
        }
    }
    __syncthreads();
    // Stage W2 over W1 in LDS.
    for (int i = tid; i < D * (D / 4); i += 256) {
        int row = i / (D / 4), col4 = i % (D / 4);
        float4 v = *reinterpret_cast<const float4*>(W2 + row * D + col4 * 4);
        *reinterpret_cast<float4*>(&w_lds[row * WS + col4 * 4]) = v;
    }
    __syncthreads();

    // GEMM2: out = hidden @ W2 + b2
    {
        const int n0 = wave * 16;
        const float bias = b2[n0 + ln15];
        for (int mt = 0; mt < 4; ++mt) {
            v8f acc = {};
            const float* arow = &h_lds[(mt * 16 + ln15) * ZS + 2 * kh];
            const float* brow = &w_lds[(2 * kh) * WS + n0 + ln15];
            for (int k = 0; k < D; k += 4) {
                v2f a, b;
                a.x = arow[k];          a.y = arow[k + 1];
                b.x = brow[k * WS];     b.y = brow[k * WS + WS];
                acc = wmma_f32_4(a, b, acc);
            }
            for (int r = 0; r < 8; ++r) {
                int row = m_block + mt * 16 + r + kh * 8;
                if (row < n_nodes) out[(size_t)row * D + n0 + ln15] = acc[r] + bias;
            }
        }
    }
}

// ------------------------------------ logits = h @ Wc + bc (128 -> 16) -------
__global__ __launch_bounds__(256) void gin_classifier(const float* __restrict__ h,
                                                      const float* __restrict__ Wc,
                                                      const float* __restrict__ bc,
                                                      float* __restrict__ logits,
                                                      int n_nodes) {
    __shared__ float wc_lds[D * WCS];
    const int tid  = threadIdx.x;
    const int wave = tid >> 5;
    const int lane = tid & 31;
    const int ln15 = lane & 15;
    const int kh   = lane >> 4;

    for (int i = tid; i < D * 16; i += 256) {
        int row = i >> 4, col = i & 15;
        wc_lds[row * WCS + col] = Wc[i];
    }
    __syncthreads();

    int m0 = blockIdx.x * 128 + wave * 16;
    if (m0 >= n_nodes) return;
    int row_ld = m0 + ln15;
    if (row_ld >= n_nodes) row_ld = n_nodes - 1;   // clamp loads; stores are guarded

    v8f acc = {};
    const float* arow = h + (size_t)row_ld * D + 2 * kh;
    const float* brow = &wc_lds[2 * kh * WCS + ln15];
    for (int k = 0; k < D; k += 4) {
        v2f a, b;
        a.x = arow[k];          a.y = arow[k + 1];
        b.x = brow[k * WCS];    b.y = brow[k * WCS + WCS];
        acc = wmma_f32_4(a, b, acc);
    }
    float bias = bc[ln15];
    for (int r = 0; r < 8; ++r) {
        int row = m0 + r + kh * 8;
        if (row < n_nodes) logits[(size_t)row * 16 + ln15] = acc[r] + bias;
    }
}

extern "C" void kernel_launch(void* const* d_in, const int* in_sizes, int n_in,
                              void* d_out, int out_size, void* d_ws, size_t ws_size,
                              hipStream_t stream) {
    const float* x  = (const float*)d_in[0];
    const int*   ei = (const int*)d_in[1];
    const float* W1 = (const float*)d_in[2];
    const float* b1 = (const float*)d_in[3];
    const float* W2 = (const float*)d_in[4];
    const float* b2 = (const float*)d_in[5];
    const float* Wc = (const float*)d_in[6];
    const float* bc = (const float*)d_in[7];

    const int n_nodes  = in_sizes[0] / D;
    const int n_edges  = in_sizes[1] / 2;
    const int n_layers = in_sizes[2] / (D * D);

    float* out_h      = (float*)d_out;                    // [n_nodes, 128]
    float* out_logits = out_h + (size_t)n_nodes * D;      // [n_nodes, 16]

    // Workspace layout: z | rowptr(n+1) | cursor(n) | col(ne)
    const size_t zbytes = (size_t)n_nodes * D * sizeof(float);
    char* w = (char*)d_ws;
    float* z      = (float*)w;
    int*   rowptr = (int*)(w + zbytes);
    int*   cursor = rowptr + (n_nodes + 1);
    int*   col    = cursor + n_nodes;
    const size_t need = zbytes + (size_t)(2 * n_nodes + 1 + n_edges) * sizeof(int);
    const bool use_csr = (ws_size >= need);

    const int* src = ei;
    const int* dst = ei + n_edges;

    if (use_csr) {
        // Build dst-indexed CSR once; reused by all layers (gather, zero f32 atomics).
        zero_i32<<<(n_nodes + 255) / 256, 256, 0, stream>>>(rowptr, n_nodes);
        count_deg<<<(n_edges + 255) / 256, 256, 0, stream>>>(dst, rowptr, n_edges);
        // rowptr currently holds degrees; scan converts in place (reads before writes).
        scan_rowptr<<<1, 256, 0, stream>>>(/*deg=*/rowptr, /*rowptr=*/rowptr,
                                           /*cursor=*/cursor, n_nodes);
        fill_csr<<<(n_edges + 255) / 256, 256, 0, stream>>>(src, dst, cursor, col, n_edges);
    }

    const float* h_cur = x;
    for (int l = 0; l < n_layers; ++l) {
        if (use_csr) {
            const long threads = (long)n_nodes * 32;
            gin_aggregate<<<(int)((threads + 255) / 256), 256, 0, stream>>>(
                h_cur, rowptr, col, z, n_nodes);
        } else {
            const int n4 = n_nodes * (D / 4);
            copy_f4<<<(n4 + 255) / 256, 256, 0, stream>>>((const float4*)h_cur,
                                                          (float4*)z, n4);
            const long threads = (long)n_edges * 32;
            gin_scatter<<<(int)((threads + 255) / 256), 256, 0, stream>>>(h_cur, src, dst,
                                                                          z, n_edges);
        }
        gin_mlp<<<(n_nodes + TILE_M - 1) / TILE_M, 256, 0, stream>>>(
            z, W1 + (size_t)l * D * D, b1 + (size_t)l * D,
            W2 + (size_t)l * D * D, b2 + (size_t)l * D, out_h, n_nodes);
        h_cur = out_h;  // MLP reads only z, so writing h in-place is safe
    }

    gin_classifier<<<(n_nodes + 127) / 128, 256, 0, stream>>>(out_h, Wc, bc, out_logits,
                                                              n_nodes);
}